// BailingAttention_80522046865495
// MI455X (gfx1250) — compile-verified
//
#include <hip/hip_runtime.h>
#include <hip/hip_bf16.h>
#include <math.h>

typedef __attribute__((ext_vector_type(2))) float v2f;
typedef __attribute__((ext_vector_type(8))) float v8f;

#define T_    2048
#define HID_  2048
#define H_    16
#define KH_   4
#define D_    128
#define QKV_N 3072   // (H + 2K) * D

// ---------------------------------------------------------------------------
// CDNA5 async global->LDS copy (ASYNCcnt path). Per-lane 16B transfer.
// Generic pointers to __shared__ have the LDS byte offset in their low 32 bits
// (flat LDS aperture maps addr[31:0] -> LDS), so truncation gives VDST.
// ---------------------------------------------------------------------------
__device__ __forceinline__ void async_ld_b128(void* lds_ptr, const void* gptr) {
  uint32_t loff = (uint32_t)(uintptr_t)lds_ptr;
  asm volatile("global_load_async_to_lds_b128 %0, %1, off"
               :: "v"(loff), "v"(gptr) : "memory");
}
__device__ __forceinline__ void async_wait0() {
  asm volatile("s_wait_asynccnt 0x0" ::: "memory");
}

// ---------------------------------------------------------------------------
// GEMM: C[M,N] = A[M,Kd] * B[Kd,N], fp32 row-major, V_WMMA_F32_16X16X4_F32.
// 256 threads = 8 waves (4 in M x 2 in N). Wave tile 32x32 (4 accumulators),
// block tile 128x64. K staged in LDS chunks of 16, DOUBLE-BUFFERED and filled
// with GLOBAL_LOAD_ASYNC_TO_LDS_B128: chunk k+1 streams into buf^1 while WMMAs
// consume buf -> one barrier per chunk, loads hidden under matrix ops.
// WMMA f32 16x16x4 fragment layout (ISA 7.12.2):
//   A: lane L<16 -> {A[M=L][kk], A[M=L][kk+1]}, lane>=16 -> kk+2/kk+3
//   B: lane L<16 -> {B[kk][N=L], B[kk+1][N=L]}, lane>=16 -> kk+2/kk+3
//   C: vgpr j -> row j (lanes 0-15) / row j+8 (lanes 16-31), col = lane&15
// ---------------------------------------------------------------------------
__global__ __launch_bounds__(256)
void gemm_f32_wmma(const float* __restrict__ A, const float* __restrict__ B,
                   float* __restrict__ C, int M, int N, int Kd) {
  __shared__ float As[2][128][20];  // pitch 20: conflict-free col reads, 16B rows
  __shared__ float Bs[2][16][68];   // pitch 68: 16B-aligned rows

  const int tid  = threadIdx.x;
  const int lane = tid & 31;
  const int w    = tid >> 5;
  const int wm   = w & 3;        // 0..3 (M direction)
  const int wn   = w >> 2;       // 0..1 (N direction)
  const int bm   = blockIdx.y * 128;
  const int bn   = blockIdx.x * 64;
  const int mr   = lane & 15;
  const int hi   = lane >> 4;    // 0 or 1

  v8f acc[2][2] = {};

  const int arow = tid >> 1;         // 0..127
  const int acol = (tid & 1) * 8;    // 0 or 8  (two b128 each)
  const int brow = tid >> 4;         // 0..15
  const int bcol = (tid & 15) * 4;   // 0..60   (one b128 each)

  // issue async fills for a K-chunk into buffer `buf`
  auto stage = [&](int k0, int buf) {
    const float* ap = A + (size_t)(bm + arow) * Kd + k0 + acol;
    async_ld_b128(&As[buf][arow][acol],     ap);
    async_ld_b128(&As[buf][arow][acol + 4], ap + 4);
    const float* bp = B + (size_t)(k0 + brow) * N + bn + bcol;
    async_ld_b128(&Bs[buf][brow][bcol], bp);
  };

  const int nk = Kd >> 4;
  stage(0, 0);

  for (int c = 0; c < nk; ++c) {
    const int buf = c & 1;
    async_wait0();     // my fills for `buf` complete (per-wave) ...
    __syncthreads();   // ... and everyone else's too; prior reads of buf done
    if (c + 1 < nk) stage((c + 1) << 4, buf ^ 1);  // overlap next fill w/ WMMA

#pragma unroll
    for (int kk = 0; kk < 16; kk += 4) {
      const int kc = kk + (hi << 1);
      v2f a0, a1, b0, b1;
      a0.x = As[buf][wm * 32 + mr][kc];      a0.y = As[buf][wm * 32 + mr][kc + 1];
      a1.x = As[buf][wm * 32 + 16 + mr][kc]; a1.y = As[buf][wm * 32 + 16 + mr][kc + 1];
      b0.x = Bs[buf][kc][wn * 32 + mr];      b0.y = Bs[buf][kc + 1][wn * 32 + mr];
      b1.x = Bs[buf][kc][wn * 32 + 16 + mr]; b1.y = Bs[buf][kc + 1][wn * 32 + 16 + mr];
      acc[0][0] = __builtin_amdgcn_wmma_f32_16x16x4_f32(false, a0, false, b0,
                                                        (short)0, acc[0][0], false, false);
      acc[0][1] = __builtin_amdgcn_wmma_f32_16x16x4_f32(false, a0, false, b1,
                                                        (short)0, acc[0][1], false, false);
      acc[1][0] = __builtin_amdgcn_wmma_f32_16x16x4_f32(false, a1, false, b0,
                                                        (short)0, acc[1][0], false, false);
      acc[1][1] = __builtin_amdgcn_wmma_f32_16x16x4_f32(false, a1, false, b1,
                                                        (short)0, acc[1][1], false, false);
    }
  }

  // ---- store C ----
#pragma unroll
  for (int mi = 0; mi < 2; ++mi)
#pragma unroll
    for (int ni = 0; ni < 2; ++ni)
#pragma unroll
      for (int j = 0; j < 8; ++j) {
        const int row = bm + wm * 32 + mi * 16 + j + (hi << 3);
        const int col = bn + wn * 32 + ni * 16 + mr;
        C[(size_t)row * N + col] = acc[mi][ni][j];
      }
}

// ---------------------------------------------------------------------------
// NeoX RoPE applied in-place to q ([T,H,D]) and k ([T,KH,D]) slices of qkv.
// ---------------------------------------------------------------------------
__global__ void rope_kernel(float* __restrict__ qkv, const int* __restrict__ pos_ids) {
  const int total = T_ * (H_ + KH_) * (D_ / 2);
  int idx = blockIdx.x * blockDim.x + threadIdx.x;
  if (idx >= total) return;
  const int i  = idx & 63;
  const int hh = (idx >> 6) % (H_ + KH_);
  const int t  = idx / (64 * (H_ + KH_));
  const int col = (hh < H_) ? hh * D_ : (H_ * D_ + (hh - H_) * D_);
  float* p = qkv + (size_t)t * QKV_N + col;
  const float inv = __expf(-((float)(2 * i) / (float)D_) * 9.210340371976184f);
  const float ang = (float)pos_ids[t] * inv;
  const float c = __cosf(ang), s = __sinf(ang);
  const float x1 = p[i], x2 = p[i + 64];
  p[i]      = x1 * c - x2 * s;
  p[i + 64] = x2 * c + x1 * s;
}

// ---------------------------------------------------------------------------
// Causal GQA flash attention. Block = 4 waves (128 thr): wave w owns 16 q-rows,
// block owns 64 q-rows of one head. K/V 16x128 blocks streamed via async-to-LDS.
// Online softmax with per-row m/l in registers (8 rows/lane in C layout);
// P converted C-layout -> A-layout through per-wave LDS tile.
// ---------------------------------------------------------------------------
__global__ __launch_bounds__(128)
void attn_kernel(const float* __restrict__ qkv, float* __restrict__ ctx) {
  __shared__ float Qs[64][132];   // pitch 132 -> bank stride 4, conflict-free
  __shared__ float Ks[16][132];
  __shared__ float Vs[16][132];
  __shared__ float Ps[4][16][20];

  const int tid  = threadIdx.x;
  const int lane = tid & 31;
  const int w    = tid >> 5;
  const int mr   = lane & 15;
  const int hi   = lane >> 4;
  const int h    = blockIdx.y;
  const int kvh  = h >> 2;
  const int q0   = blockIdx.x * 64;

  // ---- load Q tile 64x128 (this head, rope'd) ----
  {
    const int r     = tid >> 1;
    const int cbase = (tid & 1) * 64;
    const float* src = qkv + (size_t)(q0 + r) * QKV_N + h * D_ + cbase;
#pragma unroll
    for (int c = 0; c < 64; c += 4)
      *(float4*)&Qs[r][cbase + c] = *(const float4*)(src + c);
  }

  v8f acc[8] = {};
  float m_[8], l_[8];
#pragma unroll
  for (int j = 0; j < 8; ++j) { m_[j] = -3.0e38f; l_[j] = 0.0f; }

  const int jbmax = (q0 >> 4) + 3;
  const float scale = 0.08838834764831845f;  // 1/sqrt(128)

  const int kr = tid >> 3;          // 0..15
  const int cc = (tid & 7) * 16;    // 0..112, 16 floats (4 x b128) per thread

  for (int jb = 0; jb <= jbmax; ++jb) {
    __syncthreads();  // previous iteration's K/V reads complete
    { // ---- async-stage K and V blocks 16x128 directly into LDS ----
      const float* kp = qkv + (size_t)(jb * 16 + kr) * QKV_N + H_ * D_ + kvh * D_ + cc;
      const float* vp = kp + KH_ * D_;
#pragma unroll
      for (int c = 0; c < 16; c += 4) {
        async_ld_b128(&Ks[kr][cc + c], kp + c);
        async_ld_b128(&Vs[kr][cc + c], vp + c);
      }
    }
    async_wait0();
    __syncthreads();

    // ---- S(16x16) = Q * K^T over D=128 (32 WMMAs) ----
    v8f S = {};
#pragma unroll
    for (int kk = 0; kk < D_; kk += 4) {
      const int kc = kk + (hi << 1);
      v2f a, b;
      a.x = Qs[w * 16 + mr][kc]; a.y = Qs[w * 16 + mr][kc + 1];
      b.x = Ks[mr][kc];          b.y = Ks[mr][kc + 1];   // B[d][n] = K[n][d]
      S = __builtin_amdgcn_wmma_f32_16x16x4_f32(false, a, false, b,
                                                (short)0, S, false, false);
    }

    // ---- scale + causal mask + online softmax update ----
#pragma unroll
    for (int j = 0; j < 8; ++j) {
      const int grow = q0 + w * 16 + j + (hi << 3);
      const int gcol = jb * 16 + mr;
      float s = S[j] * scale;
      if (gcol > grow) s = -1.0e30f;
      float mx = s;
      mx = fmaxf(mx, __shfl_xor(mx, 1, 32));
      mx = fmaxf(mx, __shfl_xor(mx, 2, 32));
      mx = fmaxf(mx, __shfl_xor(mx, 4, 32));
      mx = fmaxf(mx, __shfl_xor(mx, 8, 32));   // reduce over 16-lane half
      const float mnew = fmaxf(m_[j], mx);
      const float fac  = __expf(m_[j] - mnew);
      const float p    = __expf(s - mnew);
      m_[j] = mnew;
      float rs = p;
      rs += __shfl_xor(rs, 1, 32);
      rs += __shfl_xor(rs, 2, 32);
      rs += __shfl_xor(rs, 4, 32);
      rs += __shfl_xor(rs, 8, 32);
      l_[j] = l_[j] * fac + rs;
#pragma unroll
      for (int n = 0; n < 8; ++n) acc[n][j] *= fac;
      Ps[w][j + (hi << 3)][mr] = p;  // C-layout -> LDS (per wave, DS in-order)
    }

    // ---- acc += P * V (A = P 16x16, B = V 16x128, 32 WMMAs) ----
#pragma unroll
    for (int kk = 0; kk < 16; kk += 4) {
      const int kc = kk + (hi << 1);
      v2f a;
      a.x = Ps[w][mr][kc]; a.y = Ps[w][mr][kc + 1];
#pragma unroll
      for (int n = 0; n < 8; ++n) {
        v2f b;
        b.x = Vs[kc][n * 16 + mr]; b.y = Vs[kc + 1][n * 16 + mr];
        acc[n] = __builtin_amdgcn_wmma_f32_16x16x4_f32(false, a, false, b,
                                                       (short)0, acc[n], false, false);
      }
    }
  }

  // ---- epilogue: ctx[q, h*D + d] = acc / l ----
#pragma unroll
  for (int n = 0; n < 8; ++n)
#pragma unroll
    for (int j = 0; j < 8; ++j) {
      const int row = q0 + w * 16 + j + (hi << 3);
      const int col = n * 16 + mr;
      ctx[(size_t)row * (H_ * D_) + h * D_ + col] = acc[n][j] / l_[j];
    }
}

// ---------------------------------------------------------------------------
extern "C" void kernel_launch(void* const* d_in, const int* in_sizes, int n_in,
                              void* d_out, int out_size, void* d_ws, size_t ws_size,
                              hipStream_t stream) {
  const float* hidden = (const float*)d_in[0];   // [T, HID]
  const int*   pos    = (const int*)d_in[1];     // [T]
  const float* Wqkv   = (const float*)d_in[2];   // [HID, 3072]
  const float* Wo     = (const float*)d_in[3];   // [H*D, HID]
  float*       out    = (float*)d_out;           // [T, HID]

  float* qkv  = (float*)d_ws;                    // [T, 3072]
  float* ctxb = qkv + (size_t)T_ * QKV_N;        // [T, 2048]

  gemm_f32_wmma<<<dim3(QKV_N / 64, T_ / 128), 256, 0, stream>>>(
      hidden, Wqkv, qkv, T_, QKV_N, HID_);

  const int nrope = T_ * (H_ + KH_) * (D_ / 2);
  rope_kernel<<<(nrope + 255) / 256, 256, 0, stream>>>(qkv, pos);

  attn_kernel<<<dim3(T_ / 64, H_), 128, 0, stream>>>(qkv, ctxb);

  gemm_f32_wmma<<<dim3(HID_ / 64, T_ / 128), 256, 0, stream>>>(
      ctxb, Wo, out, T_, HID_, H_ * D_);
}